// MVProp_39273180954735
// MI455X (gfx1250) — compile-verified
//
#include <hip/hip_runtime.h>
#include <math.h>

#define BIMG 64
#define HH 192
#define WW 192
#define NPX (HH * WW)          // 36864 pixels per image (147456 B as f32)
#define KITERS 35
#define NEGINF (-3.0e38f)
#define TPB_ITER 512           // 16 waves of 32

#define F4(p)  (*reinterpret_cast<float4*>(p))
#define CF4(p) (*reinterpret_cast<const float4*>(p))

// ---------------------------------------------------------------------------
// CDNA5 async global -> LDS copy (ASYNCcnt path), 16B per lane. GVS mode:
// saddr = 64-bit SGPR pair, vaddr = 32-bit per-lane byte offset.
// ---------------------------------------------------------------------------
__device__ __forceinline__ void async_ld_b128(unsigned lds_byte_addr,
                                              const void* saddr,
                                              unsigned voffset) {
  asm volatile("global_load_async_to_lds_b128 %0, %1, %2"
               :
               : "v"(lds_byte_addr), "v"(voffset), "s"(saddr)
               : "memory");
}

__device__ __forceinline__ void wait_asynccnt0() {
  asm volatile("s_wait_asynccnt 0" ::: "memory");
}

// Load 14-wide haloed row slice [c0-1, c0+12] from an LDS plane.
// Middle 12 are 16B-aligned (c0 = 12*tx) -> 3x ds_load_b128.
__device__ __forceinline__ void load_row14(const float* __restrict__ plane,
                                           int r, int c0, float* dst,
                                           float padv) {
  if (r < 0 || r >= HH) {
#pragma unroll
    for (int j = 0; j < 14; ++j) dst[j] = padv;
    return;
  }
  const float* row = plane + r * WW;
  float4 m0 = CF4(row + c0);
  float4 m1 = CF4(row + c0 + 4);
  float4 m2 = CF4(row + c0 + 8);
  dst[1] = m0.x;  dst[2] = m0.y;  dst[3] = m0.z;  dst[4] = m0.w;
  dst[5] = m1.x;  dst[6] = m1.y;  dst[7] = m1.z;  dst[8] = m1.w;
  dst[9] = m2.x;  dst[10] = m2.y; dst[11] = m2.z; dst[12] = m2.w;
  dst[0]  = (c0 == 0) ? padv : row[c0 - 1];
  dst[13] = (c0 + 12 >= WW) ? padv : row[c0 + 12];
}

// Load 12-wide aligned row slice [c0, c0+11] from an LDS plane.
__device__ __forceinline__ void load_row12(const float* __restrict__ plane,
                                           int r, int c0, float* dst,
                                           float padv) {
  if (r < 0 || r >= HH) {
#pragma unroll
    for (int j = 0; j < 12; ++j) dst[j] = padv;
    return;
  }
  const float* row = plane + r * WW + c0;
  float4 m0 = CF4(row);
  float4 m1 = CF4(row + 4);
  float4 m2 = CF4(row + 8);
  dst[0] = m0.x; dst[1] = m0.y; dst[2]  = m0.z; dst[3]  = m0.w;
  dst[4] = m1.x; dst[5] = m1.y; dst[6]  = m1.z; dst[7]  = m1.w;
  dst[8] = m2.x; dst[9] = m2.y; dst[10] = m2.z; dst[11] = m2.w;
}

__device__ __forceinline__ void store_row12(float* __restrict__ dstp,
                                            const float* src) {
  float4 q;
  q.x = src[0]; q.y = src[1]; q.z = src[2]; q.w = src[3];  F4(dstp) = q;
  q.x = src[4]; q.y = src[5]; q.z = src[6]; q.w = src[7];  F4(dstp + 4) = q;
  q.x = src[8]; q.y = src[9]; q.z = src[10]; q.w = src[11]; F4(dstp + 8) = q;
}

// ---------------------------------------------------------------------------
// One block per image. LDS: bufA[NPX] + bufB[NPX] = 294912 B.
// Pipeline: grid --async--> bufA; h=relu(conv1(bufA)) -> bufB;
//           p=sigmoid(conv2(bufB)) -> pout, a -> abuf, v0=goal-1 -> bufA;
//           35x { hmax(bufA) -> bufB ; v=max(v, a + p*vmax(bufB)) in bufA }.
// Thread tile: 6 rows x 12 cols (32 x 16 thread grid).
// ---------------------------------------------------------------------------
__global__ __launch_bounds__(TPB_ITER) void mvprop_fused(
    const float* __restrict__ X,
    const float* __restrict__ hp_w, const float* __restrict__ hp_b,
    const float* __restrict__ p_w,  const float* __restrict__ p_b,
    float* __restrict__ vout, float* __restrict__ pout,
    float* __restrict__ abuf) {
  extern __shared__ float lds[];
  float* bufA = lds;        // grid, then v
  float* bufB = lds + NPX;  // h, then hmax scratch

  const int img = blockIdx.x;
  const int tid = threadIdx.x;
  const int tx = tid & 15;   // 12-wide column band
  const int ty = tid >> 4;   // 6-tall row band
  const int c0 = tx * 12;
  const int r0 = ty * 6;

  const float* grid = X + (size_t)img * (2 * NPX);
  const float* goal = grid + NPX;
  const float* aimg = abuf + (size_t)img * NPX;
  const float* pimg = pout + (size_t)img * NPX;

  // uniform weight loads (scalar path)
  float w1[9], w2[9];
#pragma unroll
  for (int k = 0; k < 9; ++k) { w1[k] = hp_w[k]; w2[k] = p_w[k]; }
  const float b1v = hp_b[0], b2v = p_b[0];

  // ---- (a) async-load grid plane into bufA ----
  {
    unsigned basA = (unsigned)(uintptr_t)bufA;  // low 32 bits = LDS offset
    for (int j = tid; j < NPX / 4; j += TPB_ITER)
      async_ld_b128(basA + (unsigned)j * 16u, (const void*)grid,
                    (unsigned)j * 16u);
    wait_asynccnt0();
  }
  __syncthreads();

  // ---- (b) h = relu(conv3x3(bufA)) -> bufB (zero padding) ----
  {
    float rA[14], rB[14], rC[14];
    load_row14(bufA, r0 - 1, c0, rA, 0.0f);
    load_row14(bufA, r0,     c0, rB, 0.0f);
#pragma unroll
    for (int rr = 0; rr < 6; ++rr) {
      load_row14(bufA, r0 + rr + 1, c0, rC, 0.0f);
      float out[12];
#pragma unroll
      for (int j = 0; j < 12; ++j) {
        float s = b1v;
        s = fmaf(rA[j], w1[0], s); s = fmaf(rA[j+1], w1[1], s); s = fmaf(rA[j+2], w1[2], s);
        s = fmaf(rB[j], w1[3], s); s = fmaf(rB[j+1], w1[4], s); s = fmaf(rB[j+2], w1[5], s);
        s = fmaf(rC[j], w1[6], s); s = fmaf(rC[j+1], w1[7], s); s = fmaf(rC[j+2], w1[8], s);
        out[j] = fmaxf(s, 0.0f);
      }
      store_row12(bufB + (r0 + rr) * WW + c0, out);
#pragma unroll
      for (int j = 0; j < 14; ++j) { rA[j] = rB[j]; rB[j] = rC[j]; }
    }
  }
  __syncthreads();

  // ---- (c) p = sigmoid(conv3x3(bufB)); write p, a; seed v in bufA ----
  {
    float rA[14], rB[14], rC[14];
    load_row14(bufB, r0 - 1, c0, rA, 0.0f);
    load_row14(bufB, r0,     c0, rB, 0.0f);
#pragma unroll
    for (int rr = 0; rr < 6; ++rr) {
      const int r = r0 + rr;
      load_row14(bufB, r + 1, c0, rC, 0.0f);
      float pv[12], av[12], vv[12];
      const float* gl = goal + r * WW + c0;
      float4 g0 = CF4(gl), g1 = CF4(gl + 4), g2 = CF4(gl + 8);
      float go[12] = {g0.x,g0.y,g0.z,g0.w, g1.x,g1.y,g1.z,g1.w,
                      g2.x,g2.y,g2.z,g2.w};
#pragma unroll
      for (int j = 0; j < 12; ++j) {
        float s = b2v;
        s = fmaf(rA[j], w2[0], s); s = fmaf(rA[j+1], w2[1], s); s = fmaf(rA[j+2], w2[2], s);
        s = fmaf(rB[j], w2[3], s); s = fmaf(rB[j+1], w2[4], s); s = fmaf(rB[j+2], w2[5], s);
        s = fmaf(rC[j], w2[6], s); s = fmaf(rC[j+1], w2[7], s); s = fmaf(rC[j+2], w2[8], s);
        float p = 1.0f / (1.0f + __expf(-s));
        float rr_ = go[j] - 1.0f;          // r = goal - 1
        pv[j] = p;
        av[j] = rr_ * (1.0f - p);          // a = r * (1 - p)
        vv[j] = rr_;                        // v0 = r
      }
      store_row12((float*)(pimg + r * WW + c0), pv);   // output p (global)
      store_row12((float*)(aimg + r * WW + c0), av);   // workspace a (global)
      store_row12(bufA + r * WW + c0, vv);             // v seed (LDS, own px)
#pragma unroll
      for (int j = 0; j < 14; ++j) { rA[j] = rB[j]; rB[j] = rC[j]; }
    }
  }
  __syncthreads();

  // ---- (d) 35 Jacobi iterations entirely in LDS ----
  for (int it = 0; it < KITERS; ++it) {
    // Phase 1: horizontal 3-max for my 6 rows -> bufB
#pragma unroll
    for (int rr = 0; rr < 6; ++rr) {
      const int r = r0 + rr;
      float cur[14], out[12];
      load_row14(bufA, r, c0, cur, NEGINF);
#pragma unroll
      for (int j = 0; j < 12; ++j)
        out[j] = fmaxf(fmaxf(cur[j], cur[j + 1]), cur[j + 2]);
      store_row12(bufB + r * WW + c0, out);
    }
    __syncthreads();

    // Phase 2: vertical 3-max over bufB + update own v rows in bufA
    float ra[12], rb[12], rc[12];
    load_row12(bufB, r0 - 1, c0, ra, NEGINF);
    load_row12(bufB, r0,     c0, rb, NEGINF);
#pragma unroll
    for (int rr = 0; rr < 6; ++rr) {
      const int r = r0 + rr;
      load_row12(bufB, r + 1, c0, rc, NEGINF);
      const int base = r * WW + c0;
      if (rr < 5) {  // prefetch next row of a/p toward the WGP
        __builtin_prefetch(aimg + base + WW, 0, 3);
        __builtin_prefetch(pimg + base + WW, 0, 3);
      }
      float av[12], pv[12], vv[12];
      load_row12(aimg, r, c0, av, 0.0f);   // global (L2-resident)
      load_row12(pimg, r, c0, pv, 0.0f);   // global (L2-resident)
      load_row12(bufA, r, c0, vv, 0.0f);   // own pixels (LDS)
      float nv[12];
#pragma unroll
      for (int j = 0; j < 12; ++j) {
        float m = fmaxf(fmaxf(ra[j], rb[j]), rc[j]);
        nv[j] = fmaxf(vv[j], fmaf(pv[j], m, av[j]));
      }
      store_row12(bufA + base, nv);        // own pixels -> no conflict
#pragma unroll
      for (int j = 0; j < 12; ++j) { ra[j] = rb[j]; rb[j] = rc[j]; }
    }
    __syncthreads();
  }

  // ---- write v out (coalesced) ----
  float* vo = vout + (size_t)img * NPX;
  for (int i = tid; i < NPX; i += TPB_ITER) vo[i] = bufA[i];
}

// ---------------------------------------------------------------------------
extern "C" void kernel_launch(void* const* d_in, const int* in_sizes, int n_in,
                              void* d_out, int out_size, void* d_ws,
                              size_t ws_size, hipStream_t stream) {
  (void)in_sizes; (void)n_in; (void)out_size; (void)ws_size;
  const float* X    = (const float*)d_in[0];  // (64,2,192,192)
  const float* hp_w = (const float*)d_in[1];  // (1,1,3,3)
  const float* hp_b = (const float*)d_in[2];  // (1,)
  const float* p_w  = (const float*)d_in[3];  // (1,1,3,3)
  const float* p_b  = (const float*)d_in[4];  // (1,)

  float* vout = (float*)d_out;                // first B*N floats: v
  float* pout = vout + (size_t)BIMG * NPX;    // second B*N floats: p
  float* abuf = (float*)d_ws;                 // B*N floats: a = r*(1-p)

  const size_t shmem = (size_t)2 * NPX * sizeof(float);  // 294912 B
  (void)hipFuncSetAttribute((const void*)mvprop_fused,
                            hipFuncAttributeMaxDynamicSharedMemorySize,
                            (int)shmem);
  mvprop_fused<<<BIMG, TPB_ITER, shmem, stream>>>(X, hp_w, hp_b, p_w, p_b,
                                                  vout, pout, abuf);
}